// KalmanFilter_76244259439242
// MI455X (gfx1250) — compile-verified
//
#include <hip/hip_runtime.h>

typedef float v2f __attribute__((ext_vector_type(2)));
typedef float v8f __attribute__((ext_vector_type(8)));

#define DEVINL __device__ __forceinline__

static constexpr int KB_ = 1024;  // batches
static constexpr int KT_ = 256;   // timesteps
static constexpr int KM_ = 4;     // measurements
static constexpr int WPB = 4;     // waves per block
static constexpr int SLOT = 1312; // floats per wave LDS region (5*256 tiles + x + resid + pad)

// ---------------------------------------------------------------------------
// Intra-wave ordering: DS ops are in-order per wave; we only need to stop the
// compiler from reordering cross-lane LDS store->load pairs, plus a dscnt wait.
DEVINL void wsync() {
  asm volatile("" ::: "memory");
  __builtin_amdgcn_wave_barrier();
  asm volatile("s_wait_dscnt 0x0" ::: "memory");
}

// D = A(16x4) x B(4x16) + C, fp32 WMMA (wave32)
DEVINL v8f wmma4(v2f a, v2f b, v8f c) {
  return __builtin_amdgcn_wmma_f32_16x16x4_f32(false, a, false, b, (short)0, c,
                                               false, false);
}

// LDS staging is column-major: element (m,n) at base[n*16 + m].
// A-layout chunk k0 of X: reg j, lane l -> X[l%16][k0 + j + 2*(l/16)]
DEVINL v2f ldA(const float* base, int k0, int m16, int h) {
  v2f a;
  a.x = base[(k0 + h) * 16 + m16];
  a.y = base[(k0 + h + 1) * 16 + m16];
  return a;
}
// B-layout chunk k0 of X: reg j, lane l -> X[k0 + j + 2*(l/16)][l%16]
DEVINL v2f ldB(const float* base, int k0, int m16, int h) {
  const float* p = base + m16 * 16 + k0 + h;
  v2f b;
  b.x = p[0];
  b.y = p[1];
  return b;
}
// Note: B-layout of X^T == A-pattern read of staged X (and vice versa).

// Store a C-layout accumulator (8 VGPRs) into LDS column-major.
DEVINL void stageC(float* base, v8f c, int m16, int hi) {
  float* p = base + m16 * 16 + hi * 8;
  *(float4*)(p + 0) = make_float4(c[0], c[1], c[2], c[3]);
  *(float4*)(p + 4) = make_float4(c[4], c[5], c[6], c[7]);
}

// Full 16x16x16 matmuls (4 WMMA each), operand sources vary:
DEVINL v8f mm_AA(const float* A, const float* B, v8f c, int m16, int h) {
#pragma unroll
  for (int k0 = 0; k0 < 16; k0 += 4)
    c = wmma4(ldA(A, k0, m16, h), ldB(B, k0, m16, h), c);
  return c;
}
DEVINL v8f mm_RB(const v2f* Ar, const float* B, v8f c, int m16, int h) {
#pragma unroll
  for (int k0 = 0; k0 < 16; k0 += 4)
    c = wmma4(Ar[k0 >> 2], ldB(B, k0, m16, h), c);
  return c;
}
DEVINL v8f mm_AR(const float* A, const v2f* Br, v8f c, int m16, int h) {
#pragma unroll
  for (int k0 = 0; k0 < 16; k0 += 4)
    c = wmma4(ldA(A, k0, m16, h), Br[k0 >> 2], c);
  return c;
}
// B operand is the transpose of the staged matrix Bt.
DEVINL v8f mm_ATB(const float* A, const float* Bt, v8f c, int m16, int h) {
#pragma unroll
  for (int k0 = 0; k0 < 16; k0 += 4)
    c = wmma4(ldA(A, k0, m16, h), ldA(Bt, k0, m16, h), c);
  return c;
}

// P <- F P F^T + Q.  Fa is F in A-layout == F^T in B-layout (same registers).
DEVINL void predictP(float* sP, float* sT, const v2f* Fa, v8f Qc, int m16,
                     int h, int hi) {
  v8f z = {};
  v8f fp = mm_RB(Fa, sP, z, m16, h);  // F * P
  stageC(sT, fp, m16, hi);
  wsync();
  v8f pp = mm_AR(sT, Fa, Qc, m16, h);  // (FP) * F^T + Q
  stageC(sP, pp, m16, hi);
  wsync();
}

__global__ __launch_bounds__(WPB * 32) void kf_kernel(
    const float* __restrict__ inp, const float* __restrict__ F,
    const float* __restrict__ H, const float* __restrict__ Q,
    const float* __restrict__ R, const float* __restrict__ x0,
    const float* __restrict__ sd, float* __restrict__ out) {
  __shared__ float lds[WPB][SLOT];

  const int lane = threadIdx.x & 31;
  const int warp = threadIdx.x >> 5;
  const int b = blockIdx.x * WPB + warp;
  const int m16 = lane & 15;
  const int hi = lane >> 4;  // half-wave index
  const int h = hi << 1;     // K offset used by A/B layouts

  float* sP = &lds[warp][0];     // covariance P
  float* sT = &lds[warp][256];   // temp (HP / Pu1 / FP)
  float* sI = &lds[warp][512];   // I - K H
  float* sK = &lds[warp][768];   // -K, padded 16x16 (cols 4..15 stay zero)
  float* sS = &lds[warp][1024];  // S4 / T3
  float* sX = &lds[warp][1280];  // state x (16)
  float* sR = &lds[warp][1296];  // residual (4)

// zero the K pad region once (cols 4..15 are never written again)
#pragma unroll
  for (int i = 0; i < 8; ++i) sK[lane * 8 + i] = 0.0f;

  // ---- constant operand layouts (loaded once, live in VGPRs) ----
  v2f Fa[4], Ha[4];
#pragma unroll
  for (int c0 = 0; c0 < 4; ++c0) {
    Fa[c0].x = F[m16 * 16 + 4 * c0 + h];
    Fa[c0].y = F[m16 * 16 + 4 * c0 + h + 1];
    if (m16 < 4) {  // Hpad: rows 0..3 = H, rest zero (A-layout of Hpad)
      Ha[c0].x = H[m16 * 16 + 4 * c0 + h];
      Ha[c0].y = H[m16 * 16 + 4 * c0 + h + 1];
    } else {
      Ha[c0].x = 0.0f;
      Ha[c0].y = 0.0f;
    }
  }
  v2f HB;  // chunk 0 of Hpad in B-layout (rows 0..3 of H)
  HB.x = H[h * 16 + m16];
  HB.y = H[(h + 1) * 16 + m16];
  v2f RA;  // chunk 0 of Rpad in A-layout
  if (m16 < 4) {
    RA.x = R[m16 * 4 + h];
    RA.y = R[m16 * 4 + h + 1];
  } else {
    RA.x = 0.0f;
    RA.y = 0.0f;
  }
  v8f Qc, Rc, Ic;  // C-layout constants
#pragma unroll
  for (int r = 0; r < 8; ++r) {
    int mr = r + hi * 8;
    Qc[r] = Q[mr * 16 + m16];
    Rc[r] = (mr < 4 && m16 < 4) ? R[mr * 4 + m16] : 0.0f;
    Ic[r] = (mr == m16) ? 1.0f : 0.0f;
  }
  float Frow[16], Hrow[16];  // per-lane rows for the x-vector chain
#pragma unroll
  for (int j = 0; j < 16; ++j) {
    Frow[j] = F[m16 * 16 + j];        // row s = lane%16 of F
    Hrow[j] = H[(lane & 3) * 16 + j]; // row m = lane%4  of H
  }

  // ---- initial state: x0, P0 = diag(sd^2), then one predict ----
  {
    float s = sd[m16];
    v8f P0;
#pragma unroll
    for (int r = 0; r < 8; ++r) {
      int mr = r + hi * 8;
      P0[r] = (mr == m16) ? s * s : 0.0f;
    }
    stageC(sP, P0, m16, hi);
    if (lane < 16) sX[lane] = x0[lane];
    wsync();
    predictP(sP, sT, Fa, Qc, m16, h, hi);
    float xv[16];
#pragma unroll
    for (int j = 0; j < 16; ++j) xv[j] = sX[j];
    float xp = 0.0f;
#pragma unroll
    for (int j = 0; j < 16; ++j) xp = fmaf(Frow[j], xv[j], xp);
    wsync();
    if (lane < 16) sX[lane] = xp;
    wsync();
  }

  const float* inB = inp + (size_t)b * KT_ * KM_;
  float* outB = out + (size_t)b * KT_ * KM_;
  v8f z = {};

#pragma unroll 1
  for (int t = 0; t < KT_; ++t) {
    // ---- emit out[b,t,:] = H @ x (prior mean) ----
    float xv[16];
#pragma unroll
    for (int j = 0; j < 16; ++j) xv[j] = sX[j];
    float hx = 0.0f;
#pragma unroll
    for (int j = 0; j < 16; ++j) hx = fmaf(Hrow[j], xv[j], hx);
    if (lane < 4) outB[t * KM_ + lane] = hx;
    if (t == KT_ - 1) break;

    float obs = inB[t * KM_ + (lane & 3)];
    float resid = obs - hx;  // per-lane role m = lane%4

    // ---- covariance measurement update ----
    v8f HP = mm_RB(Ha, sP, z, m16, h);  // Hpad @ P (rows 0..3 valid)
    stageC(sT, HP, m16, hi);
    wsync();
    v8f S4 = mm_AR(sT, Ha, Rc, m16, h);  // HP @ Hpad^T + Rpad (4x4 valid)
    stageC(sS, S4, m16, hi);
    wsync();

    // per-lane 4x4 solve: S4 * y = HP[:, lane%16]  ->  y = K[lane%16, :]
    float sm[4][4];
#pragma unroll
    for (int c = 0; c < 4; ++c) {
      float4 col = *(const float4*)(sS + c * 16);
      sm[0][c] = col.x;
      sm[1][c] = col.y;
      sm[2][c] = col.z;
      sm[3][c] = col.w;
    }
    float4 rhs = *(const float4*)(sT + m16 * 16);
    float bb[4] = {rhs.x, rhs.y, rhs.z, rhs.w};
#pragma unroll
    for (int i = 0; i < 4; ++i) {
      float inv = 1.0f / sm[i][i];
      sm[i][i] = inv;  // keep inverse for back-substitution
#pragma unroll
      for (int r = i + 1; r < 4; ++r) {
        float f = sm[r][i] * inv;
#pragma unroll
        for (int c = i + 1; c < 4; ++c) sm[r][c] = fmaf(-f, sm[i][c], sm[r][c]);
        bb[r] = fmaf(-f, bb[i], bb[r]);
      }
    }
    float kk[4];
#pragma unroll
    for (int i = 3; i >= 0; --i) {
      float acc = bb[i];
#pragma unroll
      for (int c = i + 1; c < 4; ++c) acc = fmaf(-sm[i][c], kk[c], acc);
      kk[i] = acc * sm[i][i];
    }
    wsync();
    if (lane < 16) {  // stage -K (cols 0..3 of padded 16x16)
      sK[0 * 16 + lane] = -kk[0];
      sK[1 * 16 + lane] = -kk[1];
      sK[2 * 16 + lane] = -kk[2];
      sK[3 * 16 + lane] = -kk[3];
    }
    wsync();

    // ImKH = (-K)pad @ Hpad + I        (single chunk: K cols 4..15 are zero)
    v8f ImKH = wmma4(ldA(sK, 0, m16, h), HB, Ic);
    stageC(sI, ImKH, m16, hi);
    wsync();
    // Pu1 = ImKH @ P
    v8f Pu1 = mm_AA(sI, sP, z, m16, h);
    stageC(sT, Pu1, m16, hi);
    wsync();
    // T3 = Rpad @ (-K)^T = -(R K^T)    (single chunk)
    v8f T3 = wmma4(RA, ldA(sK, 0, m16, h), z);
    stageC(sS, T3, m16, hi);
    wsync();
    // KRK = (-K)pad @ T3 = K R K^T     (single chunk)
    v8f KRK = wmma4(ldA(sK, 0, m16, h), ldB(sS, 0, m16, h), z);
    // P = Pu1 @ ImKH^T + K R K^T   (Joseph form)
    v8f Pn = mm_ATB(sT, sI, KRK, m16, h);
    stageC(sP, Pn, m16, hi);
    wsync();

    // ---- state update: x += K @ resid ----
    if (lane < 4) sR[lane] = resid;
    wsync();
    float4 rr = *(const float4*)sR;
    float xs = sX[m16];
    float xu = xs + (kk[0] * rr.x + kk[1] * rr.y + kk[2] * rr.z + kk[3] * rr.w);
    wsync();
    if (lane < 16) sX[lane] = xu;
    wsync();

    // ---- time update (predict) ----
    predictP(sP, sT, Fa, Qc, m16, h, hi);
    float xuv[16];
#pragma unroll
    for (int j = 0; j < 16; ++j) xuv[j] = sX[j];
    float xp = 0.0f;
#pragma unroll
    for (int j = 0; j < 16; ++j) xp = fmaf(Frow[j], xuv[j], xp);
    wsync();
    if (lane < 16) sX[lane] = xp;
    wsync();
  }
}

extern "C" void kernel_launch(void* const* d_in, const int* in_sizes, int n_in,
                              void* d_out, int out_size, void* d_ws,
                              size_t ws_size, hipStream_t stream) {
  const float* inp = (const float*)d_in[0];
  const float* F = (const float*)d_in[1];
  const float* H = (const float*)d_in[2];
  const float* Q = (const float*)d_in[3];
  const float* R = (const float*)d_in[4];
  const float* x0 = (const float*)d_in[5];
  const float* sd = (const float*)d_in[6];
  float* out = (float*)d_out;

  dim3 grid(KB_ / WPB), block(WPB * 32);
  kf_kernel<<<grid, block, 0, stream>>>(inp, F, H, Q, R, x0, sd, out);
}